// TAGConvN_1451698946529
// MI455X (gfx1250) — compile-verified
//
#include <hip/hip_runtime.h>

typedef __attribute__((ext_vector_type(2))) float v2f;
typedef __attribute__((ext_vector_type(8))) float v8f;

#define NNODES 100000
#define NEDGES 1600000
#define DFEAT  128      // IN == H == 128
#define DC     40
#define DCPAD  48
#define KCAT   384      // (K+1)*128

// ---------------- utility kernels ----------------

__global__ void zero_f32(float* __restrict__ p, long n) {
    long i = (long)blockIdx.x * blockDim.x + threadIdx.x;
    if (i < n) p[i] = 0.f;
}

__global__ void deg_accum(const int* __restrict__ dst, float* __restrict__ deg, int nE) {
    int i = blockIdx.x * blockDim.x + threadIdx.x;
    if (i < nE) atomicAdd(&deg[dst[i]], 1.0f);
}

__global__ void norm_transform(float* __restrict__ norm, int n) {
    int i = blockIdx.x * blockDim.x + threadIdx.x;
    if (i < n) {
        float d = norm[i];
        norm[i] = (d > 0.f) ? (1.0f / sqrtf(d)) : 0.f;
    }
}

// push-SpMM: one wave per edge, 128 features = 4 per lane (float4)
// out[dst] += norm[src] * H[src]   (features stay L2-resident: 51MB < 192MB)
__global__ void spmm_push(const float* __restrict__ H, const int* __restrict__ src,
                          const int* __restrict__ dst, const float* __restrict__ norm,
                          float* __restrict__ out, int nE) {
    int gid  = blockIdx.x * blockDim.x + threadIdx.x;
    int e    = gid >> 5;
    int lane = threadIdx.x & 31;
    if (e >= nE) return;                       // wave-uniform exit
    int s = src[e], d = dst[e];
    float ns = norm[s];
    float4 v = ((const float4*)(H + (size_t)s * DFEAT))[lane];
    float* o = out + (size_t)d * DFEAT + lane * 4;
    atomicAdd(o + 0, v.x * ns);
    atomicAdd(o + 1, v.y * ns);
    atomicAdd(o + 2, v.z * ns);
    atomicAdd(o + 3, v.w * ns);
}

// post-scatter scale: h[i,f] *= norm[i]
__global__ void scale_rows(float* __restrict__ h, const float* __restrict__ norm, long n) {
    long i = (long)blockIdx.x * blockDim.x + threadIdx.x;
    if (i < n) h[i] *= norm[i >> 7];           // D = 128
}

// pad W2 (384x40) -> Wp (384x48) with zero columns so GEMM tile loads need no guard
__global__ void pad_w2(const float* __restrict__ W2, float* __restrict__ Wp) {
    int i = blockIdx.x * blockDim.x + threadIdx.x;
    if (i < KCAT * DCPAD) {
        int k = i / DCPAD, j = i % DCPAD;
        Wp[i] = (j < DC) ? W2[k * DC + j] : 0.f;
    }
}

// ---------------- fused 3-source WMMA GEMM ----------------
// Y[m, n] = relu( sum_s X_s[m, :] @ W[s*128 : (s+1)*128, n] + bias[n] )
// fp32 WMMA (V_WMMA_F32_16X16X4_F32) keeps fp32 reference precision.
// Block: 256 threads = 8 waves; each wave owns a 16x16 tile; block covers 128 rows x 16 cols.
// W panel (384x16 = 24KB) staged in LDS, shared by all 8 waves.
__global__ void tag_gemm3_wmma(const float* __restrict__ X0, const float* __restrict__ X1,
                               const float* __restrict__ X2, const float* __restrict__ W,
                               const float* __restrict__ bias, float* __restrict__ Y,
                               int nRows, int ldw, int ldy, int nCols) {
    __shared__ float wlds[KCAT * 16];
    const int tid     = threadIdx.x;
    const int colBase = blockIdx.y * 16;

    for (int i = tid; i < KCAT * 16; i += 256) {
        int k = i >> 4, j = i & 15;
        wlds[i] = W[k * ldw + colBase + j];    // W is padded, always in-bounds
    }
    __syncthreads();

    const int wave   = tid >> 5;
    const int lane   = tid & 31;
    const int laneHi = lane >> 4;              // selects K pair {0,1} vs {2,3}
    const int laneLo = lane & 15;              // M (for A) / N (for B,C)
    const int m0     = blockIdx.x * 128 + wave * 16;

    int mrow = m0 + laneLo;                    // clamp (not predicate): EXEC all-1s for WMMA
    if (mrow > nRows - 1) mrow = nRows - 1;

    v8f c = {0.f, 0.f, 0.f, 0.f, 0.f, 0.f, 0.f, 0.f};

    #pragma unroll 1
    for (int s2 = 0; s2 < 3; ++s2) {
        const float* S    = (s2 == 0) ? X0 : ((s2 == 1) ? X1 : X2);
        const float* srow = S + (size_t)mrow * DFEAT;
        const float* wk   = wlds + (s2 * 128) * 16;
        #pragma unroll 4
        for (int kk = 0; kk < 128; kk += 4) {
            int ka = kk + 2 * laneHi;
            // A 16x4 f32: lane<16 -> K={kk,kk+1}, lane>=16 -> K={kk+2,kk+3}
            v2f a = *(const v2f*)(srow + ka);
            // B 4x16 f32: VGPR0 row K=kk (+2 for hi half), VGPR1 next row
            v2f b;
            b.x = wk[(ka    ) * 16 + laneLo];
            b.y = wk[(ka + 1) * 16 + laneLo];
            c = __builtin_amdgcn_wmma_f32_16x16x4_f32(
                    /*neg_a=*/false, a, /*neg_b=*/false, b,
                    /*c_mod=*/(short)0, c, /*reuse_a=*/false, /*reuse_b=*/false);
        }
    }

    // C/D layout: VGPR r -> M = m0 + r + 8*laneHi, N = colBase + laneLo
    int n = colBase + laneLo;
    if (n < nCols) {
        float bv = bias[n];
        #pragma unroll
        for (int r = 0; r < 8; ++r) {
            int m = m0 + r + 8 * laneHi;
            if (m < nRows) {
                float v = c[r] + bv;
                Y[(size_t)m * ldy + n] = fmaxf(v, 0.f);
            }
        }
    }
}

// ---------------- mean over nodes ----------------

__global__ void mean_reduce(const float* __restrict__ Y, float* __restrict__ acc, int nRows) {
    float local[DC];
    #pragma unroll
    for (int cIdx = 0; cIdx < DC; ++cIdx) local[cIdx] = 0.f;
    for (int i = blockIdx.x * blockDim.x + threadIdx.x; i < nRows;
         i += gridDim.x * blockDim.x) {
        const float* row = Y + (size_t)i * DC;
        #pragma unroll
        for (int cIdx = 0; cIdx < DC; ++cIdx) local[cIdx] += row[cIdx];
    }
    #pragma unroll
    for (int cIdx = 0; cIdx < DC; ++cIdx) atomicAdd(&acc[cIdx], local[cIdx]);
}

__global__ void finalize_mean(const float* __restrict__ acc, float* __restrict__ out) {
    int c = threadIdx.x;
    if (c < DC) out[c] = acc[c] * (1.0f / (float)NNODES);
}

// ---------------- launch ----------------

extern "C" void kernel_launch(void* const* d_in, const int* in_sizes, int n_in,
                              void* d_out, int out_size, void* d_ws, size_t ws_size,
                              hipStream_t stream) {
    const float* x   = (const float*)d_in[0];  // [N,128]
    const int*   src = (const int*)  d_in[1];  // [E]
    const int*   dst = (const int*)  d_in[2];  // [E]
    const float* W1  = (const float*)d_in[3];  // [384,128]
    const float* b1  = (const float*)d_in[4];  // [128]
    const float* W2  = (const float*)d_in[5];  // [384,40]
    const float* b2  = (const float*)d_in[6];  // [40]
    float* out = (float*)d_out;                // [40]

    // workspace layout (floats): ~170 MB total
    float* ws   = (float*)d_ws;
    size_t off  = 0;
    float* norm = ws + off; off += NNODES;
    float* A1   = ws + off; off += (size_t)NNODES * DFEAT;
    float* A2   = ws + off; off += (size_t)NNODES * DFEAT;
    float* Y1   = ws + off; off += (size_t)NNODES * DFEAT;
    float* Y2   = ws + off; off += (size_t)NNODES * DC;
    float* Wp   = ws + off; off += (size_t)KCAT * DCPAD;
    float* acc  = ws + off; off += 64;

    const int TPB = 256;
    const long featN      = (long)NNODES * DFEAT;
    const int  featBlocks = (int)((featN + TPB - 1) / TPB);
    const int  spmmBlocks = (NEDGES * 32 + TPB - 1) / TPB;
    const int  nodeBlocks = (NNODES + TPB - 1) / TPB;
    const int  edgeBlocks = (NEDGES + TPB - 1) / TPB;

    // degree -> D^{-1/2}
    zero_f32<<<nodeBlocks, TPB, 0, stream>>>(norm, NNODES);
    deg_accum<<<edgeBlocks, TPB, 0, stream>>>(dst, norm, NEDGES);
    norm_transform<<<nodeBlocks, TPB, 0, stream>>>(norm, NNODES);

    // layer 1 hops: A1 = A_hat x ; A2 = A_hat^2 x
    zero_f32<<<featBlocks, TPB, 0, stream>>>(A1, featN);
    spmm_push<<<spmmBlocks, TPB, 0, stream>>>(x, src, dst, norm, A1, NEDGES);
    scale_rows<<<featBlocks, TPB, 0, stream>>>(A1, norm, featN);
    zero_f32<<<featBlocks, TPB, 0, stream>>>(A2, featN);
    spmm_push<<<spmmBlocks, TPB, 0, stream>>>(A1, src, dst, norm, A2, NEDGES);
    scale_rows<<<featBlocks, TPB, 0, stream>>>(A2, norm, featN);

    // layer 1 GEMM + relu  (M=100000, K=384, N=128)
    dim3 g1((NNODES + 127) / 128, DFEAT / 16);
    tag_gemm3_wmma<<<g1, TPB, 0, stream>>>(x, A1, A2, W1, b1, Y1,
                                           NNODES, DFEAT, DFEAT, DFEAT);

    // layer 2 hops (reuse A1/A2): A1 = A_hat y1 ; A2 = A_hat^2 y1
    zero_f32<<<featBlocks, TPB, 0, stream>>>(A1, featN);
    spmm_push<<<spmmBlocks, TPB, 0, stream>>>(Y1, src, dst, norm, A1, NEDGES);
    scale_rows<<<featBlocks, TPB, 0, stream>>>(A1, norm, featN);
    zero_f32<<<featBlocks, TPB, 0, stream>>>(A2, featN);
    spmm_push<<<spmmBlocks, TPB, 0, stream>>>(A1, src, dst, norm, A2, NEDGES);
    scale_rows<<<featBlocks, TPB, 0, stream>>>(A2, norm, featN);

    // layer 2 GEMM + relu (cols padded 40 -> 48 for clean 16-wide tiles)
    pad_w2<<<(KCAT * DCPAD + TPB - 1) / TPB, TPB, 0, stream>>>(W2, Wp);
    dim3 g2((NNODES + 127) / 128, DCPAD / 16);
    tag_gemm3_wmma<<<g2, TPB, 0, stream>>>(Y1, A1, A2, Wp, b2, Y2,
                                           NNODES, DCPAD, DC, DC);

    // mean over nodes -> [1, 40]
    zero_f32<<<1, 64, 0, stream>>>(acc, 64);
    mean_reduce<<<200, TPB, 0, stream>>>(Y2, acc, NNODES);
    finalize_mean<<<1, 64, 0, stream>>>(acc, out);
}